// DebugModule_65566970741221
// MI455X (gfx1250) — compile-verified
//
#include <hip/hip_runtime.h>
#include <math.h>

// Problem constants (match reference)
#define NL      12
#define NC      4
#define GD      2
#define BATCH   4
#define LEN     64
#define TSTEPS  (LEN - 1)                 // 63, second product uses cs[:, :-1]
#define CS_ELEMS  (NL*NL*NC*NC*GD)        // 4608  (A,J,k,m,g)
#define CS_PAIRS  (NL*NL*NC*NC)           // 2304  (A,J,k,m)
#define TP_ELEMS  (NL*NL*NC*GD)           // 1152  (J,C,c3,g)
#define PAIRS_PER_CS (NL*NC)              // 48    (C,c3)
#define PAIR_TOTAL  (CS_PAIRS*PAIRS_PER_CS) // 110592, divisible by 256
#define BCAST_N  (BATCH*LEN*TP_ELEMS)     // 294912 per tensor

#define LOG_2PI_F 1.8378770664093453f

typedef float v2f __attribute__((ext_vector_type(2)));
typedef float v8f __attribute__((ext_vector_type(8)));

// Lane-xor butterfly step via DS_SWIZZLE_B32 group-of-32 mode:
// offset = (xor_mask << 10) | (or_mask=0 << 5) | (and_mask=0x1f)
// Template parameter keeps the offset an ICE for the builtin.
template <int OFFS>
__device__ __forceinline__ float xor_swizzle_add(float s) {
    int v = __builtin_amdgcn_ds_swizzle(__float_as_int(s), OFFS);
    return s + __int_as_float(v);
}

// Wave32 sum of (a0,a1) across all 32 lanes using the matrix unit:
// D = ones(16x4) * B(4x16) + 0  => every row of D holds the 16 column sums;
// the 64 B entries are exactly the wave's 2 accumulators per lane, so
// xor-reducing one D row (masks 1,2,4,8 within each 16-lane half, both
// halves hold identical rows M=0 / M=8) yields the full wave total.
__device__ __forceinline__ float wave_sum_wmma(float a0, float a1) {
    v2f ones; ones.x = 1.0f; ones.y = 1.0f;
    v2f bv;   bv.x = a0;     bv.y = a1;
    v8f c = {};
    v8f d = __builtin_amdgcn_wmma_f32_16x16x4_f32(
        /*neg_a=*/false, ones, /*neg_b=*/false, bv,
        /*c_mod=*/(short)0, c, /*reuse_a=*/false, /*reuse_b=*/false);
    float s = d[0];
    s = xor_swizzle_add<(1 << 10) | 0x1f>(s);   // SWAPX1
    s = xor_swizzle_add<(2 << 10) | 0x1f>(s);   // SWAPX2
    s = xor_swizzle_add<(4 << 10) | 0x1f>(s);   // SWAPX4
    s = xor_swizzle_add<(8 << 10) | 0x1f>(s);   // SWAPX8
    return s;  // every lane holds the wave total
}

// One workgroup per (b, t<63). Stages cs (first Gaussian product) and
// trans_p tables in LDS, then accumulates the second product's scale over
// all (A,J,k,m)x(C,c3) pairs; reduces with WMMA; writes one block partial.
__global__ __launch_bounds__(256) void csp_reduce_kernel(
    const int*   __restrict__ ids,
    const float* __restrict__ s_mu_w,  const float* __restrict__ s_var_w,
    const float* __restrict__ tc_mu,   const float* __restrict__ tc_var,
    const float* __restrict__ tp_mu,   const float* __restrict__ tp_var,
    float*       __restrict__ block_part) {

    __shared__ float sh_cs_mu[CS_ELEMS];
    __shared__ float sh_cs_w1[CS_ELEMS];
    __shared__ float sh_tp_mu[TP_ELEMS];
    __shared__ float sh_tp_w2[TP_ELEMS];
    __shared__ float sh_wave[8];

    const int tid = threadIdx.x;
    const int blk = blockIdx.x;
    const int b   = blk / TSTEPS;
    const int t   = blk - b * TSTEPS;
    const int id  = ids[b * LEN + t];
    const float* smu  = s_mu_w  + id * (NL*NC*GD);   // 96 per label
    const float* svar = s_var_w + id * (NL*NC*GD);

    // First Gaussian product: cs_mu and w1 = exp(2*cs_var) = v1*v2/(v1+v2)
    for (int i = tid; i < CS_ELEMS; i += 256) {
        const int g  = i & 1;
        const int i2 = i >> 1;          // (A,J,k,m)
        const int m  = i2 & 3;
        const int i3 = i2 >> 2;
        const int k  = i3 & 3;
        const int i4 = i3 >> 2;         // A*NL + J
        const int J  = i4 % NL;

        const float sm = smu [(J*NC + m)*GD + g];
        const float sv = svar[(J*NC + m)*GD + g];
        const int   ti = (i4*NC + k)*GD + g;     // ((A*NL+J)*NC+k)*GD+g
        const float cm = tc_mu [ti];
        const float cv = tc_var[ti];

        const float v1  = expf(2.0f * sv);
        const float v2  = expf(2.0f * cv);
        const float vs  = v1 + v2;
        sh_cs_mu[i] = (sm * v2 + cm * v1) / vs;
        sh_cs_w1[i] = (v1 * v2) / vs;            // exp(2*(sv+cv) - log(vs))
    }
    for (int i = tid; i < TP_ELEMS; i += 256) {
        sh_tp_mu[i] = tp_mu[i];
        sh_tp_w2[i] = expf(2.0f * tp_var[i]);
    }
    __syncthreads();

    // Pair loop: PAIR_TOTAL / 256 = 432 iterations per thread (uniform,
    // so EXEC stays all-ones into the WMMA below).
    float acc0 = 0.0f, acc1 = 0.0f;
    for (int p = tid; p < PAIR_TOTAL; p += 256) {
        const int csIdx = p / PAIRS_PER_CS;          // (A,J,k,m)
        const int pc    = p - csIdx * PAIRS_PER_CS;  // (C,c3)
        const int J     = (csIdx >> 4) % NL;
        const int e     = J * PAIRS_PER_CS + pc;     // ((J*NL+C)*NC+c3)

        const float mu1a = sh_cs_mu[csIdx*2 + 0];
        const float mu1b = sh_cs_mu[csIdx*2 + 1];
        const float w1a  = sh_cs_w1[csIdx*2 + 0];
        const float w1b  = sh_cs_w1[csIdx*2 + 1];
        const float mu2a = sh_tp_mu[e*2 + 0];
        const float mu2b = sh_tp_mu[e*2 + 1];
        const float w2a  = sh_tp_w2[e*2 + 0];
        const float w2b  = sh_tp_w2[e*2 + 1];

        const float wsa = w1a + w2a;
        const float wsb = w1b + w2b;
        const float da  = mu1a - mu2a;
        const float db  = mu1b - mu2b;
        acc0 += -0.5f * (LOG_2PI_F + logf(wsa) + da*da / wsa);
        acc1 += -0.5f * (LOG_2PI_F + logf(wsb) + db*db / wsb);
    }

    const float wsum = wave_sum_wmma(acc0, acc1);
    const int lane = tid & 31, wv = tid >> 5;
    if (lane == 0) sh_wave[wv] = wsum;
    __syncthreads();
    if (tid == 0) {
        float tot = 0.0f;
        #pragma unroll
        for (int i = 0; i < 8; ++i) tot += sh_wave[i];
        block_part[blk] = tot;
    }
}

// Single wave: deterministic fixed-order sum of the 252 block partials,
// final wave reduction again via WMMA.
__global__ __launch_bounds__(32) void finalize_kernel(
    const float* __restrict__ part, float* __restrict__ out) {
    float acc = 0.0f;
    for (int i = threadIdx.x; i < BATCH * TSTEPS; i += 32) acc += part[i];
    const float tot = wave_sum_wmma(acc, 0.0f);
    if (threadIdx.x == 0) out[0] = tot;
}

// Broadcast trans_p_mu / trans_p_var over (b,l): plain modulo-tile copy.
__global__ __launch_bounds__(256) void broadcast_kernel(
    const float* __restrict__ tp_mu, const float* __restrict__ tp_var,
    float* __restrict__ out) {
    const int i = blockIdx.x * 256 + threadIdx.x;
    if (i < BCAST_N) {
        out[1 + i] = tp_mu[i % TP_ELEMS];
    } else if (i < 2 * BCAST_N) {
        out[1 + i] = tp_var[(i - BCAST_N) % TP_ELEMS];
    }
}

extern "C" void kernel_launch(void* const* d_in, const int* in_sizes, int n_in,
                              void* d_out, int out_size, void* d_ws, size_t ws_size,
                              hipStream_t stream) {
    const int*   ids     = (const int*)  d_in[0];
    const float* s_mu_w  = (const float*)d_in[1];
    const float* s_var_w = (const float*)d_in[2];
    const float* tc_mu   = (const float*)d_in[3];
    const float* tc_var  = (const float*)d_in[4];
    const float* tp_mu   = (const float*)d_in[5];
    const float* tp_var  = (const float*)d_in[6];
    float* out  = (float*)d_out;
    float* part = (float*)d_ws;   // 252 floats of scratch

    csp_reduce_kernel<<<BATCH * TSTEPS, 256, 0, stream>>>(
        ids, s_mu_w, s_var_w, tc_mu, tc_var, tp_mu, tp_var, part);
    finalize_kernel<<<1, 32, 0, stream>>>(part, out);

    const int total_bcast = 2 * BCAST_N;
    broadcast_kernel<<<(total_bcast + 255) / 256, 256, 0, stream>>>(
        tp_mu, tp_var, out);
}